// MultiHeadAttention_28406913695771
// MI455X (gfx1250) — compile-verified
//
#include <hip/hip_runtime.h>
#include <hip/hip_bf16.h>

// MHA + residual + LayerNorm for MI455X (gfx1250), fp32 throughout via
// V_WMMA_F32_16X16X4_F32, with gfx1250 async global->LDS staging.
//
// Pipeline:
//   1) proj_kernel x3 : Qh/Kh/Vh = X @ W^T, head-major [B,NH,S,D] in d_ws.
//                       Tiles staged with GLOBAL_LOAD_ASYNC_TO_LDS_B128,
//                       double-buffered on ASYNCcnt.
//   2) attn_kernel    : per (b, h, 32-query block): scores (WMMA) -> mask ->
//                       softmax in LDS -> single-pass attn write -> PV (WMMA)
//                       -> +residual -> out region of d_out
//   3) ln_kernel      : in-place layernorm on out region
//
// d_out = [ out: B*S*H floats | attn: B*NH*S*S floats ]
// d_ws  = [ Qh | Kh | Vh ] each B*S*H floats (48 MB total)

#define BB 2
#define SS 2048
#define HH 1024
#define NHH 16
#define DD 64

#define SP 2052   // padded LDS stride for 32xS score strip
#define QP 68     // padded LDS stride for 32x64 Q tile

#define PBUF (128 * 33)   // one padded 128x32 tile buffer (floats)

typedef float v2f __attribute__((ext_vector_type(2)));
typedef float v8f __attribute__((ext_vector_type(8)));

__device__ __forceinline__ v8f wmma4(v2f a, v2f b, v8f c) {
  // 8 args: (neg_a, A, neg_b, B, c_mod, C, reuse_a, reuse_b)
  return __builtin_amdgcn_wmma_f32_16x16x4_f32(false, a, false, b, (short)0, c,
                                               false, false);
}

// Async copy of 16 bytes global -> LDS, tracked by ASYNCcnt. GV addressing:
// 64-bit per-lane address in a VGPR pair, LDS byte offset in a VGPR.
__device__ __forceinline__ void async_b128(void* lds, const void* gptr) {
  const unsigned int loff = (unsigned int)(uintptr_t)lds;  // LDS offset (low 32b)
  const unsigned long long ga = (unsigned long long)(uintptr_t)gptr;
  asm volatile("global_load_async_to_lds_b128 %0, %1, off"
               :
               : "v"(loff), "v"(ga)
               : "memory");
}

__device__ __forceinline__ void wait_async(int pending) {
  if (pending)
    asm volatile("s_wait_asynccnt 8" ::: "memory");
  else
    asm volatile("s_wait_asynccnt 0" ::: "memory");
}

// ---------------------------------------------------------------------------
// Projection GEMM: Y[m,n] = sum_k X[m,k] * W[n,k]   (y = x @ W^T)
// Block tile 128x128, 8 waves (4x2), wave tile 32x64 (2x4 WMMA tiles).
// Double-buffered async global->LDS staging: LDS = [A0|A1|B0|B1], each 128x33.
// ---------------------------------------------------------------------------
__global__ __launch_bounds__(256) void proj_kernel(
    const float* __restrict__ X, const float* __restrict__ W,
    float* __restrict__ Yh) {
  extern __shared__ float psm[];

  const int n0 = blockIdx.x * 128;
  const int m0 = blockIdx.y * 128;
  const int tid = threadIdx.x;
  const int wave = tid >> 5;
  const int lane = tid & 31;
  const int half = (lane >> 4) & 1;
  const int l16 = lane & 15;
  const int wm = wave >> 1;  // 0..3 : m strip = wm*32
  const int wn = wave & 1;   // 0..1 : n strip = wn*64

  v8f acc[2][4] = {};

  // each thread stages 4 float4 of A and 4 float4 of B per 32-wide K chunk
  auto issue_chunk = [&](int k0, int buf) {
    float* Ab = psm + buf * PBUF;            // A buffer 0/1
    float* Bb = psm + 2 * PBUF + buf * PBUF; // B buffer 0/1
#pragma unroll
    for (int j = 0; j < 4; ++j) {
      const int fi = tid + j * 256;  // float4 index, 0..1023
      const int row = fi >> 3;       // 8 float4 per 32-float row
      const int c4 = (fi & 7) * 4;   // stays within one padded row
      async_b128(Ab + row * 33 + c4, X + (size_t)(m0 + row) * HH + k0 + c4);
      async_b128(Bb + row * 33 + c4, W + (size_t)(n0 + row) * HH + k0 + c4);
    }
  };

  issue_chunk(0, 0);

  for (int c = 0; c < 32; ++c) {
    const int cur = c & 1;
    const float* As = psm + cur * PBUF;
    const float* Bs = psm + 2 * PBUF + cur * PBUF;
    const int more = (c < 31);
    if (more) issue_chunk((c + 1) * 32, cur ^ 1);
    wait_async(more);   // chunk c landed; chunk c+1 may stay in flight
    __syncthreads();

#pragma unroll
    for (int ks = 0; ks < 8; ++ks) {
      const int kk = ks * 4 + 2 * half;  // VGPR j <-> K = 2*half + j
      v2f af[2], bf[4];
#pragma unroll
      for (int i = 0; i < 2; ++i) {
        const int m = wm * 32 + i * 16 + l16;
        af[i] = v2f{As[m * 33 + kk], As[m * 33 + kk + 1]};
      }
#pragma unroll
      for (int j = 0; j < 4; ++j) {
        const int n = wn * 64 + j * 16 + l16;
        bf[j] = v2f{Bs[n * 33 + kk], Bs[n * 33 + kk + 1]};
      }
#pragma unroll
      for (int i = 0; i < 2; ++i)
#pragma unroll
        for (int j = 0; j < 4; ++j) acc[i][j] = wmma4(af[i], bf[j], acc[i][j]);
    }
    __syncthreads();  // reads done before this buffer is refilled (chunk c+2)
  }

  // epilogue: scatter to head-major [B,NH,S,D]
#pragma unroll
  for (int i = 0; i < 2; ++i)
#pragma unroll
    for (int j = 0; j < 4; ++j)
#pragma unroll
      for (int r = 0; r < 8; ++r) {
        const int m = m0 + wm * 32 + i * 16 + r + 8 * half;
        const int n = n0 + wn * 64 + j * 16 + l16;
        const int b = m >> 11;       // / S
        const int s = m & (SS - 1);
        const int h = n >> 6;        // / D
        const int d = n & (DD - 1);
        Yh[(((size_t)(b * NHH + h) * SS) + s) * DD + d] = acc[i][j][r];
      }
}

// ---------------------------------------------------------------------------
// Fused attention: one block = (b, h, 32 query rows).
// LDS: scores 32 x S (padded), Q tile, ctx accum, softmax scratch (~282 KB).
// ---------------------------------------------------------------------------
__global__ __launch_bounds__(256) void attn_kernel(
    const float* __restrict__ Qh, const float* __restrict__ Kh,
    const float* __restrict__ Vh, const int* __restrict__ mask,
    const float* __restrict__ qin, float* __restrict__ attn_out,
    float* __restrict__ out) {
  extern __shared__ float smem[];
  float* sc = smem;                    // [32][SP]
  float* Qs = sc + 32 * SP;            // [32][QP]
  float* ctxs = Qs + 32 * QP;          // [32][64]
  float* redA = ctxs + 32 * 64;        // [256]
  float* redB = redA + 256;            // [256]

  const int qb = blockIdx.x;           // 0..63
  const int h = blockIdx.y;            // 0..15
  const int b = blockIdx.z;            // 0..1
  const int row0 = qb * 32;

  const size_t headBase = (size_t)(b * NHH + h) * SS * DD;
  const float* Qg = Qh + headBase;
  const float* Kg = Kh + headBase;
  const float* Vg = Vh + headBase;

  const int tid = threadIdx.x;
  const int wave = tid >> 5;
  const int lane = tid & 31;
  const int half = (lane >> 4) & 1;
  const int l16 = lane & 15;

  // --- async-load Q tile (32x64), zero ctx accumulator ---
  for (int e = tid; e < 512; e += 256) {   // 512 float4s
    const int m = e >> 3;                  // 8 float4 per 64-float row
    const int c4 = (e & 7) * 4;
    async_b128(&Qs[m * QP + c4], Qg + (size_t)(row0 + m) * DD + c4);
  }
  for (int e = tid; e < 32 * DD; e += 256) ctxs[e] = 0.0f;
  asm volatile("s_wait_asynccnt 0" ::: "memory");
  __syncthreads();

  // --- Phase 1: scores[32][S] = Q K^T. Wave w owns columns [w*256, w*256+256)
  for (int nt = 0; nt < 16; ++nt) {
    v8f c0 = {}, c1 = {};
    const int ncol = wave * 256 + nt * 16 + l16;
    const float* Krow = Kg + (size_t)ncol * DD;
    if (nt < 15) __builtin_prefetch(Krow + 16 * DD, 0, 3);  // next K tile (L2)
#pragma unroll
    for (int ks = 0; ks < 16; ++ks) {
      const int kk = ks * 4 + 2 * half;
      const v2f bf = v2f{Krow[kk], Krow[kk + 1]};
      const v2f a0 = v2f{Qs[l16 * QP + kk], Qs[l16 * QP + kk + 1]};
      const v2f a1 = v2f{Qs[(16 + l16) * QP + kk], Qs[(16 + l16) * QP + kk + 1]};
      c0 = wmma4(a0, bf, c0);
      c1 = wmma4(a1, bf, c1);
    }
    const int cl = wave * 256 + nt * 16 + l16;
#pragma unroll
    for (int r = 0; r < 8; ++r) {
      sc[(r + 8 * half) * SP + cl] = c0[r];
      sc[(16 + r + 8 * half) * SP + cl] = c1[r];
    }
  }
  __syncthreads();

  // --- Phase 2: mask + softmax, write attn to HBM (single pass) ---
  {
    const int r = tid >> 3;             // row 0..31
    const int seg = tid & 7;            // 8 threads per row
    float* rowp = sc + r * SP;
    const int c0i = seg * 256;
    const int* mrow = mask + ((size_t)b * SS + row0 + r) * SS;

    float mx = -3.4e38f;
    for (int c = c0i; c < c0i + 256; ++c) {
      const float v = (mrow[c] == 0) ? -1000000000.0f : rowp[c];
      rowp[c] = v;
      mx = fmaxf(mx, v);
    }
    redA[tid] = mx;
    __syncthreads();
    float rmx = -3.4e38f;
#pragma unroll
    for (int i = 0; i < 8; ++i) rmx = fmaxf(rmx, redA[r * 8 + i]);

    float sum = 0.0f;
    for (int c = c0i; c < c0i + 256; ++c) {
      const float e = __expf(rowp[c] - rmx);
      rowp[c] = e;
      sum += e;
    }
    redB[tid] = sum;
    __syncthreads();
    float rs = 0.0f;
#pragma unroll
    for (int i = 0; i < 8; ++i) rs += redB[r * 8 + i];
    const float inv = 1.0f / rs;

    float* arow = attn_out + (((size_t)(b * NHH + h) * SS) + row0 + r) * SS;
    for (int c = c0i; c < c0i + 256; ++c) {
      const float p = rowp[c] * inv;
      rowp[c] = p;        // keep normalized probs in LDS for PV
      arow[c] = p;        // materialize attn (required output)
    }
  }
  __syncthreads();

  // --- Phase 3: ctx = P @ V. Wave w accumulates its k-strip, LDS-atomic reduce
  {
    v8f acc[2][4] = {};
    const int kbase = wave * 256;
#pragma unroll 4
    for (int ks = 0; ks < 64; ++ks) {
      const int kk = kbase + ks * 4 + 2 * half;
      const v2f a0 = v2f{sc[l16 * SP + kk], sc[l16 * SP + kk + 1]};
      const v2f a1 = v2f{sc[(16 + l16) * SP + kk], sc[(16 + l16) * SP + kk + 1]};
#pragma unroll
      for (int j = 0; j < 4; ++j) {
        const int d = j * 16 + l16;
        const v2f bf =
            v2f{Vg[(size_t)kk * DD + d], Vg[(size_t)(kk + 1) * DD + d]};
        acc[0][j] = wmma4(a0, bf, acc[0][j]);
        acc[1][j] = wmma4(a1, bf, acc[1][j]);
      }
    }
#pragma unroll
    for (int i = 0; i < 2; ++i)
#pragma unroll
      for (int j = 0; j < 4; ++j)
#pragma unroll
        for (int r = 0; r < 8; ++r) {
          const int m = i * 16 + r + 8 * half;
          const int d = j * 16 + l16;
          atomicAdd(&ctxs[m * DD + d], acc[i][j][r]);  // ds_add_f32
        }
  }
  __syncthreads();

  // --- epilogue: +residual, scatter back to token-major out [B,S,H] ---
  for (int e = tid; e < 32 * DD; e += 256) {
    const int m = e >> 6, d = e & 63;
    const size_t gi = ((size_t)b * SS + row0 + m) * HH + h * DD + d;
    out[gi] = ctxs[e] + qin[gi];
  }
}

// ---------------------------------------------------------------------------
// In-place layernorm on out region: one block per row of 1024.
// ---------------------------------------------------------------------------
__global__ __launch_bounds__(256) void ln_kernel(float* __restrict__ out,
                                                 const float* __restrict__ gamma,
                                                 const float* __restrict__ beta) {
  __shared__ float rs[256], rs2[256];
  float* x = out + (size_t)blockIdx.x * HH;
  const int tid = threadIdx.x;
  float v[4];
  float s = 0.0f, s2 = 0.0f;
#pragma unroll
  for (int i = 0; i < 4; ++i) {
    v[i] = x[tid + i * 256];
    s += v[i];
    s2 += v[i] * v[i];
  }
  rs[tid] = s;
  rs2[tid] = s2;
  __syncthreads();
  for (int off = 128; off > 0; off >>= 1) {
    if (tid < off) {
      rs[tid] += rs[tid + off];
      rs2[tid] += rs2[tid + off];
    }
    __syncthreads();
  }
  const float mu = rs[0] * (1.0f / HH);
  const float var = rs2[0] * (1.0f / HH) - mu * mu;
  const float rstd = rsqrtf(var + 1e-6f);
#pragma unroll
  for (int i = 0; i < 4; ++i) {
    const int c = tid + i * 256;
    x[c] = (v[i] - mu) * rstd * gamma[c] + beta[c];
  }
}

extern "C" void kernel_launch(void* const* d_in, const int* in_sizes, int n_in,
                              void* d_out, int out_size, void* d_ws,
                              size_t ws_size, hipStream_t stream) {
  (void)in_sizes; (void)n_in; (void)out_size; (void)ws_size;
  const float* q = (const float*)d_in[0];
  const float* k = (const float*)d_in[1];
  const float* v = (const float*)d_in[2];
  const int* mask = (const int*)d_in[3];
  const float* Wq = (const float*)d_in[4];
  const float* Wk = (const float*)d_in[5];
  const float* Wv = (const float*)d_in[6];
  const float* gamma = (const float*)d_in[7];
  const float* beta = (const float*)d_in[8];

  float* out = (float*)d_out;
  float* attn = out + (size_t)BB * SS * HH;

  float* Qh = (float*)d_ws;                       // 16 MB each
  float* Kh = Qh + (size_t)BB * SS * HH;
  float* Vh = Kh + (size_t)BB * SS * HH;

  dim3 pg(HH / 128, (BB * SS) / 128);             // (8, 32)
  const size_t pshmem = (size_t)4 * PBUF * sizeof(float);  // 67.5 KB
  proj_kernel<<<pg, 256, pshmem, stream>>>(q, Wq, Qh);
  proj_kernel<<<pg, 256, pshmem, stream>>>(k, Wk, Kh);
  proj_kernel<<<pg, 256, pshmem, stream>>>(v, Wv, Vh);

  dim3 ag(SS / 32, NHH, BB);                      // (64, 16, 2)
  const size_t ashmem =
      (size_t)(32 * SP + 32 * QP + 32 * 64 + 512) * sizeof(float);
  attn_kernel<<<ag, 256, ashmem, stream>>>(Qh, Kh, Vh, mask, q, attn, out);

  ln_kernel<<<BB * SS, 256, 0, stream>>>(out, gamma, beta);
}